// Detect_60559038874059
// MI455X (gfx1250) — compile-verified
//
#include <hip/hip_runtime.h>
#include <cstdint>

#define NC   81       // NUM_CLASSES
#define PN   16320    // priors
#define BN   16       // batch
#define TOPK 200
#define TP   256      // priors per prep block
#define NPT  16       // priors per thread in NMS (16320 <= 16*1024)

typedef float v2f __attribute__((ext_vector_type(2)));
typedef float v8f __attribute__((ext_vector_type(8)));

// ---------------------------------------------------------------------------
// Stage 1: softmax(odm_conf) with WMMA row-sums, ARM gating, box decode,
// masked score transpose (b,p,c) -> (b,c,p).  HBM-bound (~180MB).
// ---------------------------------------------------------------------------
__global__ __launch_bounds__(TP) void prep_kernel(
    const float* __restrict__ arm_loc,
    const float* __restrict__ arm_conf,
    const float* __restrict__ odm_loc,
    const float* __restrict__ odm_conf,
    const float* __restrict__ priors,
    float* __restrict__ scores_ws,   // [B][80][P]
    float* __restrict__ boxes_ws)    // [B][P][4]
{
  extern __shared__ float lds[];
  float* tile = lds;             // [TP][81], stride 81 (17 mod 64 -> conflict-free)
  float* sums = lds + TP * NC;   // [TP]

  const int tid = threadIdx.x;
  const int p0  = blockIdx.x * TP;
  const int b   = blockIdx.y;
  const int nP  = min(TP, PN - p0);

  // ---- Phase A: async-copy odm_conf tile (contiguous) into LDS ----
  {
    const int n4 = (nP * NC) >> 2;   // 256*81 and 192*81 both divisible by 4
    uint64_t src = (uint64_t)(uintptr_t)(odm_conf + ((size_t)b * PN + p0) * NC);
    uint32_t ldsbase = (uint32_t)(uintptr_t)(void*)tile; // low 32 bits = LDS offset
    for (int i = tid; i < n4; i += TP) {
      uint32_t ldst = ldsbase + (uint32_t)(i << 4);
      uint32_t voff = (uint32_t)(i << 4);
      asm volatile("global_load_async_to_lds_b128 %0, %1, %2"
                   :: "v"(ldst), "v"(voff), "s"(src) : "memory");
    }
    asm volatile("s_wait_asynccnt 0" ::: "memory");
  }
  __syncthreads();

  // ---- Phase B: per-prior max + exp (in place); ARM gate; box decode ----
  bool pos = false;
  const int p = p0 + tid;
  if (tid < nP) {
    const float2 ac = reinterpret_cast<const float2*>(arm_conf)[(size_t)b * PN + p];
    // softmax(arm)[1] > 0.01  <=>  a0 - a1 < ln(99)
    pos = (ac.x - ac.y) < 4.5951198501345898f;

    const float4 al = reinterpret_cast<const float4*>(arm_loc)[(size_t)b * PN + p];
    const float4 ol = reinterpret_cast<const float4*>(odm_loc)[(size_t)b * PN + p];
    const float4 pr = reinterpret_cast<const float4*>(priors)[p];
    float rcx = pr.x + al.x * 0.1f * pr.z;
    float rcy = pr.y + al.y * 0.1f * pr.w;
    float rsx = pr.z * __expf(al.z * 0.2f);
    float rsy = pr.w * __expf(al.w * 0.2f);
    float cx  = rcx + ol.x * 0.1f * rsx;
    float cy  = rcy + ol.y * 0.1f * rsy;
    float sx  = rsx * __expf(ol.z * 0.2f);
    float sy  = rsy * __expf(ol.w * 0.2f);
    float4 bx;
    bx.x = cx - 0.5f * sx; bx.y = cy - 0.5f * sy;
    bx.z = cx + 0.5f * sx; bx.w = cy + 0.5f * sy;
    reinterpret_cast<float4*>(boxes_ws)[(size_t)b * PN + p] = bx;

    float* row = tile + tid * NC;
    float mx = row[0];
    for (int c = 1; c < NC; ++c) mx = fmaxf(mx, row[c]);
    for (int c = 0; c < NC; ++c) row[c] = __expf(row[c] - mx);
  }
  __syncthreads();

  // ---- Phase C: softmax denominators via V_WMMA_F32_16X16X4_F32 chains ----
  // Each wave handles 2 groups of 16 priors; A = exp tile rows, B = ones.
  {
    const int lane = tid & 31;
    const int wv   = tid >> 5;
    v2f bones; bones.x = 1.0f; bones.y = 1.0f;
    for (int gi = 0; gi < 2; ++gi) {
      const int g    = wv * 2 + gi;
      const int m    = g * 16 + (lane & 15);
      const int koff = (lane >> 4) << 1;          // 0 for lanes 0-15, 2 for 16-31
      const float* row = tile + m * NC;
      v8f acc = {};
#pragma unroll
      for (int kc = 0; kc < 20; ++kc) {           // classes 0..79
        v2f a;
        a.x = row[kc * 4 + koff];
        a.y = row[kc * 4 + koff + 1];
        acc = __builtin_amdgcn_wmma_f32_16x16x4_f32(
            false, a, false, bones, (short)0, acc, false, false);
      }
      if ((lane & 15) == 0) {                     // lanes 0 (M=0..7) and 16 (M=8..15), N=0
        const int mb = g * 16 + ((lane >> 4) << 3);
#pragma unroll
        for (int r = 0; r < 8; ++r)
          sums[mb + r] = acc[r] + tile[(mb + r) * NC + 80]; // add class 80
      }
    }
  }
  __syncthreads();

  // ---- Phase D: masked, normalized, transposed score writes ----
  if (tid < nP) {
    const float inv = 1.0f / sums[tid];
    const float* row = tile + tid * NC;
    float* dst = scores_ws + (size_t)b * 80 * PN + p;
    for (int c = 1; c < NC; ++c) {
      float v = row[c] * inv;
      v = (pos && v > 0.01f) ? v : 0.0f;
      dst[(size_t)(c - 1) * PN] = v;              // coalesced along p
    }
  }
}

// ---------------------------------------------------------------------------
// Stage 2: exact greedy NMS, one 32-wave workgroup per (batch, class).
// Scores/boxes/areas in registers (16/thread); full box table in LDS (261KB)
// for the per-iteration best-box broadcast. One barrier per iteration
// (double-buffered partials). Division-free IoU test.
// ---------------------------------------------------------------------------
__global__ __launch_bounds__(1024, 1) void nms_kernel(
    const float* __restrict__ boxes_ws,
    const float* __restrict__ scores_ws,
    float* __restrict__ out)
{
  extern __shared__ float lds[];
  float4* ldsBox = reinterpret_cast<float4*>(lds);         // [P]
  float*  redS   = lds + 4 * PN;                           // [2][32]
  int*    redP   = reinterpret_cast<int*>(redS + 64);      // [2][32]

  const int tid = threadIdx.x;
  const int ci  = blockIdx.x;          // 0..79 -> output class ci+1
  const int b   = blockIdx.y;

  const float4* gBox = reinterpret_cast<const float4*>(boxes_ws) + (size_t)b * PN;
  const float*  gScr = scores_ws + ((size_t)b * 80 + ci) * PN;

  float x1[NPT], y1[NPT], x2[NPT], y2[NPT], ar[NPT], sc[NPT];
#pragma unroll
  for (int j = 0; j < NPT; ++j) {
    const int p = tid + (j << 10);
    float4 bb = make_float4(0.f, 0.f, 0.f, 0.f);
    float s = 0.f;
    if (p < PN) {
      bb = gBox[p];
      s  = gScr[p];
      ldsBox[p] = bb;
    }
    x1[j] = bb.x; y1[j] = bb.y; x2[j] = bb.z; y2[j] = bb.w;
    ar[j] = (bb.z - bb.x) * (bb.w - bb.y);
    sc[j] = s;
  }
  __syncthreads();

  const int lane = tid & 31;
  const int wv   = tid >> 5;
  float* orow0 = out + ((size_t)b * NC + (ci + 1)) * TOPK * 5;

  for (int k = 0; k < TOPK; ++k) {
    // local argmax (smallest index on ties, matching jnp.argmax)
    float bs = 0.f; int bp = 0x7fffffff;
#pragma unroll
    for (int j = 0; j < NPT; ++j) {
      const int p = tid + (j << 10);
      const bool better = (sc[j] > bs) || ((sc[j] == bs) && (p < bp));
      bs = better ? sc[j] : bs;
      bp = better ? p : bp;
    }
    // wave32 reduction
#pragma unroll
    for (int off = 16; off >= 1; off >>= 1) {
      const float os = __shfl_xor(bs, off, 32);
      const int   op = __shfl_xor(bp, off, 32);
      const bool better = (os > bs) || ((os == bs) && (op < bp));
      bs = better ? os : bs;
      bp = better ? op : bp;
    }
    const int buf = (k & 1) << 5;
    if (lane == 0) { redS[buf + wv] = bs; redP[buf + wv] = bp; }
    __syncthreads();
    // every thread folds the 32 partials (LDS broadcast reads)
    bs = redS[buf]; bp = redP[buf];
    for (int w = 1; w < 32; ++w) {
      const float os = redS[buf + w];
      const int   op = redP[buf + w];
      const bool better = (os > bs) || ((os == bs) && (op < bp));
      bs = better ? os : bs;
      bp = better ? op : bp;
    }
    if (bs <= 0.f) break;   // uniform; remaining rows stay zero (pre-memset)

    const float4 bb = ldsBox[bp];               // conflict-free broadcast
    const float  ba = (bb.z - bb.x) * (bb.w - bb.y);
    if (tid == 0) {
      float* orow = orow0 + (size_t)k * 5;
      orow[0] = bs; orow[1] = bb.x; orow[2] = bb.y; orow[3] = bb.z; orow[4] = bb.w;
    }
    // suppression: iou > T  <=>  inter > T * union (division-free)
#pragma unroll
    for (int j = 0; j < NPT; ++j) {
      const int p = tid + (j << 10);
      float iw = fminf(x2[j], bb.z) - fmaxf(x1[j], bb.x);
      float ih = fminf(y2[j], bb.w) - fmaxf(y1[j], bb.y);
      iw = fmaxf(iw, 0.f); ih = fmaxf(ih, 0.f);
      const float inter = iw * ih;
      const float uni   = ba + ar[j] - inter;
      const bool  kill  = (inter > 0.45f * uni) || (p == bp);
      sc[j] = kill ? 0.f : sc[j];
    }
  }
}

// ---------------------------------------------------------------------------
extern "C" void kernel_launch(void* const* d_in, const int* in_sizes, int n_in,
                              void* d_out, int out_size, void* d_ws, size_t ws_size,
                              hipStream_t stream) {
  const float* arm_loc  = (const float*)d_in[0];
  const float* arm_conf = (const float*)d_in[1];
  const float* odm_loc  = (const float*)d_in[2];
  const float* odm_conf = (const float*)d_in[3];
  const float* priors   = (const float*)d_in[4];
  float* out = (float*)d_out;

  // Workspace: scores [B][80][P] f32 (83.6MB) + boxes [B][P][4] f32 (4.2MB)
  float* scores_ws = (float*)d_ws;
  float* boxes_ws  = scores_ws + (size_t)BN * 80 * PN;

  // Zero output (class 0 rows + rows past early NMS termination stay zero).
  hipMemsetAsync(d_out, 0, (size_t)out_size * sizeof(float), stream);

  {
    dim3 grid((PN + TP - 1) / TP, BN);
    size_t shb = (size_t)(TP * NC + TP) * sizeof(float);   // 83,968 B
    prep_kernel<<<grid, TP, shb, stream>>>(arm_loc, arm_conf, odm_loc, odm_conf,
                                           priors, scores_ws, boxes_ws);
  }
  {
    dim3 grid(80, BN);
    size_t shb = (size_t)PN * 16 + 2 * 32 * (sizeof(float) + sizeof(int)); // 261,632 B
    nms_kernel<<<grid, 1024, shb, stream>>>(boxes_ws, scores_ws, out);
  }
}